// OnlineTripletLoss_8830452760814
// MI455X (gfx1250) — compile-verified
//
#include <hip/hip_runtime.h>
#include <hip/hip_bf16.h>
#include <math.h>

typedef float v2f __attribute__((ext_vector_type(2)));
typedef float v8f __attribute__((ext_vector_type(8)));

#define BN 8192
#define DIM 128
#define NTILE (BN / 16)        // 512 column tiles
#define LDS_STRIDE 132         // 128 + 4 pad floats -> conflict-free b64 reads
#define MARGIN 0.2f
#define FLT_BIG 3.402823466e38f
#define EPS2 1e-12f
#define SENT 0x7F800000u       // +inf bits: above any finite d2 bit pattern

// ---------------- kernel 1: per-row squared norms (warp per row) ----------------
__global__ __launch_bounds__(256) void sq_kernel(const float* __restrict__ X,
                                                 float* __restrict__ sq) {
  const int tid  = threadIdx.x;
  const int row  = blockIdx.x * 8 + (tid >> 5);
  const int lane = tid & 31;
  const float4 v = *(const float4*)(X + (size_t)row * DIM + lane * 4);
  float s = v.x * v.x + v.y * v.y + v.z * v.z + v.w * v.w;
#pragma unroll
  for (int m = 16; m >= 1; m >>= 1) s += __shfl_xor(s, m, 32);
  if (lane == 0) sq[row] = s;
}

__device__ __forceinline__ float sqrt_eps(float d2) {
  return (d2 > EPS2) ? sqrtf(d2) : 0.f;   // reference's safe-sqrt rule
}

// ---------------- kernel 2: fused Gram-tile + hard mining ----------------
// Per wave: 16-row strip, A fragment register-hoisted (WMMA f32 16x16x4 layout:
// lane<16 holds row=lane%16 K={4k,4k+1}; lane>=16 K={4k+2,4k+3}).
// B tiles double-buffered in LDS, shared by the block's 8 waves.
// Mining runs in squared-distance space, tracked as u32 bit patterns
// (d2 >= 0 -> float order == uint order; avoids maxnum canonicalize traffic).
__global__ __launch_bounds__(256) void mine_kernel(const float* __restrict__ X,
                                                   const int* __restrict__ labels,
                                                   const float* __restrict__ sq,
                                                   float* __restrict__ hp_out,
                                                   float* __restrict__ hn_out,
                                                   unsigned* __restrict__ maxbits) {
  __shared__ float tile[2][16 * LDS_STRIDE];

  const int tid     = threadIdx.x;
  const int wave    = tid >> 5;
  const int lane    = tid & 31;
  const int half    = lane >> 4;          // 0: lanes 0-15, 1: lanes 16-31
  const int n       = lane & 15;
  const int i_strip = blockIdx.x * 128 + wave * 16;

  // Hoist this wave's A strip (16 rows x 128 K) into registers, WMMA layout.
  v2f a[32];
  {
    const float* arow = X + (size_t)(i_strip + n) * DIM + 2 * half;
#pragma unroll
    for (int k = 0; k < 32; ++k) a[k] = *(const v2f*)(arow + 4 * k);
  }

  // C/D layout maps VGPR r -> row i_strip + r + 8*half, lane%16 -> column.
  float si[8]; int li[8];
#pragma unroll
  for (int r = 0; r < 8; ++r) {
    const int row = i_strip + r + 8 * half;
    si[r] = sq[row];
    li[r] = labels[row];
  }
  unsigned hp2[8], hn2[8];
#pragma unroll
  for (int r = 0; r < 8; ++r) { hp2[r] = 0u; hn2[r] = SENT; }
  unsigned maxd2 = 0u;

  // Stage one 16x128 column tile into LDS buffer `buf` (block-cooperative).
  auto stage = [&](int jt, int buf) {
    const float* src = X + (size_t)jt * 16 * DIM;
#pragma unroll
    for (int p = 0; p < 4; ++p) {
      const int q   = tid + p * 256;      // float2 index, 0..1023
      const int row = q >> 6;
      const int c2  = q & 63;
      v2f v = *(const v2f*)(src + row * DIM + c2 * 2);
      *(v2f*)&tile[buf][row * LDS_STRIDE + c2 * 2] = v;
    }
  };

  stage(0, 0);
  for (int jt = 0; jt < NTILE; ++jt) {
    __syncthreads();                      // staging of tile[jt&1] complete
    const int buf = jt & 1;
    if (jt + 1 < NTILE) stage(jt + 1, buf ^ 1);  // prefetch next, overlap compute

    // 16x16 Gram tile: 32 fp32 WMMAs over K=128, 4 independent accumulator
    // chains to hide matrix-pipe latency.
    v8f acc[4];
#pragma unroll
    for (int u = 0; u < 4; ++u) acc[u] = (v8f){0.f,0.f,0.f,0.f,0.f,0.f,0.f,0.f};
    const float* brow = &tile[buf][n * LDS_STRIDE + 2 * half];
#pragma unroll
    for (int k = 0; k < 32; ++k) {
      v2f b = *(const v2f*)(brow + 4 * k);
      acc[k & 3] = __builtin_amdgcn_wmma_f32_16x16x4_f32(false, a[k], false, b,
                                                         (short)0, acc[k & 3],
                                                         false, false);
    }
    v8f c = (acc[0] + acc[1]) + (acc[2] + acc[3]);

    // Fused epilogue: squared distance, masked u32 min/max mining.
    const int j0   = jt * 16 + n;
    const float sj = sq[j0];
    const int  lj  = labels[j0];
#pragma unroll
    for (int r = 0; r < 8; ++r) {
      const int irow = i_strip + r + 8 * half;
      const float d2f = fmaxf(fmaf(-2.f, c[r], si[r] + sj), 0.f);
      const unsigned d2 = __float_as_uint(d2f);
      const bool eq = (li[r] == lj);
      maxd2 = (d2 > maxd2) ? d2 : maxd2;
      const unsigned dp = (eq && irow != j0) ? d2 : 0u;
      hp2[r] = (dp > hp2[r]) ? dp : hp2[r];
      const unsigned dn = eq ? SENT : d2;
      hn2[r] = (dn < hn2[r]) ? dn : hn2[r];
    }
  }

  // Reduce across the 16 lanes of each half (xor masks stay within the half).
#pragma unroll
  for (int r = 0; r < 8; ++r) {
#pragma unroll
    for (int m = 8; m >= 1; m >>= 1) {
      const unsigned op = __shfl_xor(hp2[r], m, 32);
      const unsigned on = __shfl_xor(hn2[r], m, 32);
      hp2[r] = (op > hp2[r]) ? op : hp2[r];
      hn2[r] = (on < hn2[r]) ? on : hn2[r];
    }
  }
#pragma unroll
  for (int m = 16; m >= 1; m >>= 1) {
    const unsigned om = __shfl_xor(maxd2, m, 32);
    maxd2 = (om > maxd2) ? om : maxd2;
  }

  if (lane == 0) {
    const float maxd = sqrt_eps(__uint_as_float(maxd2));
    atomicMax(maxbits, __float_as_uint(maxd));  // d >= 0: uint order == float order
  }
  if (n == 0) {                                 // lanes 0 and 16 own the 16 rows
#pragma unroll
    for (int r = 0; r < 8; ++r) {
      const int row = i_strip + r + 8 * half;
      hp_out[row] = sqrt_eps(__uint_as_float(hp2[r]));
      hn_out[row] = (hn2[r] >= SENT) ? FLT_BIG
                                     : sqrt_eps(__uint_as_float(hn2[r]));
    }
  }
}

// ---------------- kernel 3: finalize (fallback + relu + mean) ----------------
__global__ __launch_bounds__(256) void finalize_kernel(const float* __restrict__ hp,
                                                       const float* __restrict__ hn,
                                                       const unsigned* __restrict__ maxbits,
                                                       float* __restrict__ out) {
  __shared__ float red[256];
  const float maxd = __uint_as_float(*maxbits);
  float s = 0.f;
  for (int r = threadIdx.x; r < BN; r += 256) {
    float neg = hn[r];
    if (neg >= FLT_BIG) neg = maxd;       // row with no negatives: min = 0 + max_dist
    s += fmaxf(hp[r] - neg + MARGIN, 0.f);
  }
  red[threadIdx.x] = s;
  __syncthreads();
  for (int m = 128; m >= 1; m >>= 1) {
    if (threadIdx.x < m) red[threadIdx.x] += red[threadIdx.x + m];
    __syncthreads();
  }
  if (threadIdx.x == 0) out[0] = red[0] / (float)BN;
}

extern "C" void kernel_launch(void* const* d_in, const int* in_sizes, int n_in,
                              void* d_out, int out_size, void* d_ws, size_t ws_size,
                              hipStream_t stream) {
  const float* X      = (const float*)d_in[0];   // embeddings [8192,128] f32
  const int*   labels = (const int*)d_in[1];     // [8192] i32
  float* out = (float*)d_out;

  float* sq = (float*)d_ws;
  float* hp = sq + BN;
  float* hn = hp + BN;
  unsigned* maxbits = (unsigned*)(hn + BN);

  hipMemsetAsync(maxbits, 0, sizeof(unsigned), stream);
  sq_kernel<<<BN / 8, 256, 0, stream>>>(X, sq);
  mine_kernel<<<BN / 128, 256, 0, stream>>>(X, labels, sq, hp, hn, maxbits);
  finalize_kernel<<<1, 256, 0, stream>>>(hp, hn, maxbits, out);
}